// AttnReadout_29566554865681
// MI455X (gfx1250) — compile-verified
//
#include <hip/hip_runtime.h>
#include <hip/hip_bf16.h>

// Problem constants from the reference
#define B_DIM   256
#define S_LEN   200
#define D_DIM   1024
#define H_DIM   1024
#define BN_EPS  1e-5f
#define NEG_BIG 2e32f

typedef __attribute__((ext_vector_type(2))) float v2f;
typedef __attribute__((ext_vector_type(8))) float v8f;

// Only meaningful during the gfx1250 device pass; host (x86) pass must skip it.
#if defined(__HIP_DEVICE_COMPILE__)
#if !__has_builtin(__builtin_amdgcn_wmma_f32_16x16x4_f32)
#error "device pass: wmma_f32_16x16x4_f32 builtin not available"
#endif
#endif

// ---------------------------------------------------------------------------
// Phase A: feat_v[b,h] = last_nodes[b,:] @ W_v[:,h] + b_v[h]
// M=256, N=1024, K=1024 fp32 WMMA. 1024 wave-tiles, 8 waves/block -> 128 blocks
// ---------------------------------------------------------------------------
__global__ __launch_bounds__(256) void gemm_v_kernel(
    const float* __restrict__ last, const float* __restrict__ W_v,
    const float* __restrict__ b_v, float* __restrict__ feat_v)
{
    const int wave = threadIdx.x >> 5;
    const int lane = threadIdx.x & 31;
    const int tileId = blockIdx.x * 8 + wave;   // 0..1023
    const int tm = tileId >> 6;                 // 16 row tiles
    const int tn = tileId & 63;                 // 64 col tiles
    const int r = lane & 15;
    const int kOff = (lane >> 4) * 2;           // A/B VGPR K split: lanes16-31 -> K+2
    const int n = tn * 16 + r;

    v8f acc = {};
    const float* ap = last + (size_t)(tm * 16 + r) * D_DIM + kOff;
    for (int k0 = 0; k0 < D_DIM; k0 += 4) {
        v2f a, bm;
        a.x = ap[k0];
        a.y = ap[k0 + 1];
        bm.x = W_v[(size_t)(k0 + kOff) * H_DIM + n];
        bm.y = W_v[(size_t)(k0 + kOff + 1) * H_DIM + n];
        acc = __builtin_amdgcn_wmma_f32_16x16x4_f32(false, a, false, bm,
                                                    (short)0, acc, false, false);
    }
    const float bias = b_v[n];
    const int mHalf = (lane >> 4) * 8;          // C layout: lanes16-31 hold rows 8..15
#pragma unroll
    for (int i = 0; i < 8; ++i) {
        const int m = tm * 16 + mHalf + i;
        feat_v[(size_t)m * H_DIM + n] = acc[i] + bias;
    }
}

// ---------------------------------------------------------------------------
// Phase B: fused BN + GEMM (x @ W_u) + sigmoid(.+feat_v) . w_e  ->  e[B*S]
// One block per 32-row tile of M = B*S = 51200 (1600 blocks).
// x-tile (32x1024, BN+mask applied) staged in LDS (~132 KB of 320 KB WGP LDS);
// each of 8 waves owns 8 of the 64 N-tiles; per K-step one B fragment feeds
// TWO wmma (row sub-tiles 0-15 and 16-31) -> halves W_u L2 traffic vs 16-row
// tiling. Lane/LDS reduce to per-row logits; deterministic (no atomics).
// ---------------------------------------------------------------------------
#define M_TILE   32
#define XS_PITCH 1028   // 1024 + 4 pad floats -> conflict-free ds reads

__global__ __launch_bounds__(256) void fused_logits_kernel(
    const float* __restrict__ feat, const float* __restrict__ mask,
    const float* __restrict__ bn_gamma, const float* __restrict__ bn_beta,
    const float* __restrict__ bn_mean, const float* __restrict__ bn_var,
    const float* __restrict__ W_u, const float* __restrict__ feat_v,
    const float* __restrict__ w_e, float* __restrict__ e_out)
{
    __shared__ float xs[M_TILE * XS_PITCH];
    __shared__ float rowScale[M_TILE];
    __shared__ float rowShift[M_TILE];
    __shared__ float partial[8][M_TILE];

    const int tid  = threadIdx.x;
    const int wave = tid >> 5;
    const int lane = tid & 31;
    const int rowBase = blockIdx.x * M_TILE;  // global M base, M = b*S + s

    // Per-row BN constants with mask folded in: x = feat*sc + sh
    if (tid < M_TILE) {
        const int m  = rowBase + tid;
        const int ss = m % S_LEN;
        const float inv = rsqrtf(bn_var[ss] + BN_EPS);
        const float sc  = bn_gamma[ss] * inv;
        const float sh  = bn_beta[ss] - bn_mean[ss] * sc;
        const float mk  = mask[m];
        rowScale[tid] = sc * mk;
        rowShift[tid] = sh * mk;
    }
    __syncthreads();

    // Stage normalized x tile: 32 rows x 1024, float4 loads (32 iters/thread)
    for (int v = tid; v < (M_TILE * D_DIM) / 4; v += 256) {
        const int row = v >> 8;                 // 256 float4 per row
        const int k4  = (v & 255) * 4;
        const float4 f = *(const float4*)(feat + (size_t)(rowBase + row) * D_DIM + k4);
        const float sc = rowScale[row], sh = rowShift[row];
        float* dst = xs + row * XS_PITCH + k4;
        dst[0] = f.x * sc + sh;
        dst[1] = f.y * sc + sh;
        dst[2] = f.z * sc + sh;
        dst[3] = f.w * sc + sh;
    }
    __syncthreads();

    const int r     = lane & 15;
    const int kOff  = (lane >> 4) * 2;
    const int mHalf = (lane >> 4) * 8;
    const float* ap0 = xs + r * XS_PITCH + kOff;          // rows 0..15
    const float* ap1 = xs + (16 + r) * XS_PITCH + kOff;   // rows 16..31

    float gAcc0[8], gAcc1[8];
#pragma unroll
    for (int i = 0; i < 8; ++i) { gAcc0[i] = 0.f; gAcc1[i] = 0.f; }

    for (int t = 0; t < 8; ++t) {
        const int n = (wave * 8 + t) * 16 + r;
        const float we = w_e[n];
        v8f acc0 = {};
        v8f acc1 = {};
#pragma unroll 4
        for (int k0 = 0; k0 < D_DIM; k0 += 4) {
            v2f a0, a1, bm;
            a0.x = ap0[k0];                     // ds_load_b64: x[r][k0+kOff..+1]
            a0.y = ap0[k0 + 1];
            a1.x = ap1[k0];                     // ds_load_b64: x[16+r][...]
            a1.y = ap1[k0 + 1];
            bm.x = W_u[(size_t)(k0 + kOff) * H_DIM + n];
            bm.y = W_u[(size_t)(k0 + kOff + 1) * H_DIM + n];
            if ((k0 & 63) == 0)                 // global_prefetch_b8 of B stream
                __builtin_prefetch(W_u + (size_t)(k0 + kOff + 64) * H_DIM + n, 0, 1);
            acc0 = __builtin_amdgcn_wmma_f32_16x16x4_f32(false, a0, false, bm,
                                                         (short)0, acc0, false, false);
            acc1 = __builtin_amdgcn_wmma_f32_16x16x4_f32(false, a1, false, bm,
                                                         (short)0, acc1, false, false);
        }
        // epilogue: sigmoid(u + feat_v[b, n]) * w_e[n], accumulate per row
#pragma unroll
        for (int i = 0; i < 8; ++i) {
            const int m0  = rowBase + mHalf + i;
            const int m1  = m0 + 16;
            const int bb0 = m0 / S_LEN;
            const int bb1 = m1 / S_LEN;
            const float u0 = acc0[i] + feat_v[(size_t)bb0 * H_DIM + n];
            const float u1 = acc1[i] + feat_v[(size_t)bb1 * H_DIM + n];
            gAcc0[i] += we / (1.f + __expf(-u0));
            gAcc1[i] += we / (1.f + __expf(-u1));
        }
    }

    // Reduce over the 16 N-lanes within each half (halves hold disjoint rows)
#pragma unroll
    for (int i = 0; i < 8; ++i) {
        float g0 = gAcc0[i];
        float g1 = gAcc1[i];
        for (int off = 8; off >= 1; off >>= 1) {
            g0 += __shfl_xor(g0, off, 16);
            g1 += __shfl_xor(g1, off, 16);
        }
        if (r == 0) {
            partial[wave][mHalf + i]      = g0;
            partial[wave][16 + mHalf + i] = g1;
        }
    }
    __syncthreads();

    if (tid < M_TILE) {
        float s = 0.f;
#pragma unroll
        for (int w = 0; w < 8; ++w) s += partial[w][tid];
        e_out[rowBase + tid] = s;
    }
}

// ---------------------------------------------------------------------------
// Phase C1: masked softmax over S per batch -> beta[B*S]
// ---------------------------------------------------------------------------
__global__ __launch_bounds__(256) void softmax_beta_kernel(
    const float* __restrict__ e, const float* __restrict__ mask,
    float* __restrict__ beta)
{
    __shared__ float red[256];
    const int b = blockIdx.x;
    const int t = threadIdx.x;

    float ev = -3.0e38f;
    if (t < S_LEN)
        ev = e[b * S_LEN + t] + (mask[b * S_LEN + t] - 1.f) * NEG_BIG;

    red[t] = ev;
    __syncthreads();
    for (int off = 128; off; off >>= 1) {
        if (t < off) red[t] = fmaxf(red[t], red[t + off]);
        __syncthreads();
    }
    const float mx = red[0];
    __syncthreads();

    const float ex = (t < S_LEN) ? __expf(ev - mx) : 0.f;
    red[t] = ex;
    __syncthreads();
    for (int off = 128; off; off >>= 1) {
        if (t < off) red[t] += red[t + off];
        __syncthreads();
    }
    const float inv = 1.f / red[0];
    if (t < S_LEN) beta[b * S_LEN + t] = ex * inv;
}

// ---------------------------------------------------------------------------
// Phase C2: rst[b,d] = sum_s x[b,s,d] * beta[b,s]   (memory-bound stream)
// grid = (B, D/256); BN constants + beta cached in LDS per block.
// ---------------------------------------------------------------------------
__global__ __launch_bounds__(256) void weighted_sum_kernel(
    const float* __restrict__ feat, const float* __restrict__ mask,
    const float* __restrict__ bn_gamma, const float* __restrict__ bn_beta,
    const float* __restrict__ bn_mean, const float* __restrict__ bn_var,
    const float* __restrict__ beta, float* __restrict__ out)
{
    __shared__ float sc[S_LEN], sh[S_LEN], bw[S_LEN];
    const int b = blockIdx.x;
    const int d = blockIdx.y * 256 + threadIdx.x;
    const int t = threadIdx.x;

    if (t < S_LEN) {
        const float inv = rsqrtf(bn_var[t] + BN_EPS);
        const float s   = bn_gamma[t] * inv;
        const float mk  = mask[b * S_LEN + t];
        sc[t] = s * mk;
        sh[t] = (bn_beta[t] - bn_mean[t] * s) * mk;
        bw[t] = beta[b * S_LEN + t];
    }
    __syncthreads();

    float acc = 0.f;
    const float* fp = feat + (size_t)b * S_LEN * D_DIM + d;
    for (int s = 0; s < S_LEN; ++s)
        acc = fmaf(fp[(size_t)s * D_DIM] * sc[s] + sh[s], bw[s], acc);

    out[(size_t)b * D_DIM + d] = acc;
}

// ---------------------------------------------------------------------------
// Host launcher
// ---------------------------------------------------------------------------
extern "C" void kernel_launch(void* const* d_in, const int* in_sizes, int n_in,
                              void* d_out, int out_size, void* d_ws, size_t ws_size,
                              hipStream_t stream) {
    const float* feat     = (const float*)d_in[0];
    const float* last     = (const float*)d_in[1];
    const float* mask     = (const float*)d_in[2];
    const float* bn_gamma = (const float*)d_in[3];
    const float* bn_beta  = (const float*)d_in[4];
    const float* bn_mean  = (const float*)d_in[5];
    const float* bn_var   = (const float*)d_in[6];
    const float* W_u      = (const float*)d_in[7];
    const float* W_v      = (const float*)d_in[8];
    const float* b_v      = (const float*)d_in[9];
    const float* w_e      = (const float*)d_in[10];
    float* out = (float*)d_out;

    float* ws     = (float*)d_ws;
    float* feat_v = ws;                              // B*H   = 262144 floats
    float* e_buf  = feat_v + (size_t)B_DIM * H_DIM;  // B*S   =  51200 floats
    float* beta   = e_buf + (size_t)B_DIM * S_LEN;   // B*S   =  51200 floats

    // A: feat_v = last @ W_v + b_v       (1024 wave-tiles / 8 waves per block)
    gemm_v_kernel<<<128, 256, 0, stream>>>(last, W_v, b_v, feat_v);

    // B: fused BN + x@W_u + sigmoid-reduce -> e   (51200/32 = 1600 row tiles)
    fused_logits_kernel<<<1600, 256, 0, stream>>>(feat, mask, bn_gamma, bn_beta,
                                                  bn_mean, bn_var, W_u, feat_v,
                                                  w_e, e_buf);

    // C1: masked softmax over S
    softmax_beta_kernel<<<B_DIM, 256, 0, stream>>>(e_buf, mask, beta);

    // C2: weighted sum over S -> rst[B, D]
    dim3 gridC(B_DIM, D_DIM / 256);
    weighted_sum_kernel<<<gridC, 256, 0, stream>>>(feat, mask, bn_gamma, bn_beta,
                                                   bn_mean, bn_var, beta, out);
}